// WideAndDeepModel_71700184039630
// MI455X (gfx1250) — compile-verified
//
#include <hip/hip_runtime.h>
#include <hip/hip_bf16.h>
#include <math.h>

// ---------------------------------------------------------------------------
// Wide & Deep forward, MI455X (gfx1250, wave32, WMMA).
//
// Key algebraic rewrite: one_hot(items) @ w_head[:, :10000].T is a scalar
// gather w_head[j, items[b]] -- never materialize the one-hot (saves ~819
// GFLOP of zero-multiplies and 160 MB of one-hot traffic).
//
// All GEMMs run on V_WMMA_F32_16X16X4_F32 (exact f32, memory-bound problem).
// ---------------------------------------------------------------------------

#if defined(__HIP_DEVICE_COMPILE__)
#if !__has_builtin(__builtin_amdgcn_wmma_f32_16x16x4_f32)
#error "gfx1250 device pass: __builtin_amdgcn_wmma_f32_16x16x4_f32 not available"
#endif
#endif

typedef __attribute__((ext_vector_type(2))) float v2f;
typedef __attribute__((ext_vector_type(8))) float v8f;

#define BATCH     4096
#define NUM_ITEMS 10000
#define EMB_DIM   32
#define D_IN      64
#define H1        128
#define H2        128
#define HEAD_IN   (NUM_ITEMS + H2)   // 10128

// ---------------------------------------------------------------------------
// 1) Embedding gather + concat: embed[b, 0:32]=users_emb[u], [32:64]=items_emb[i]
// ---------------------------------------------------------------------------
__global__ void gather_embed_kernel(const int* __restrict__ inter,
                                    const float* __restrict__ users_emb,
                                    const float* __restrict__ items_emb,
                                    float* __restrict__ embed) {
  const int b = blockIdx.x;        // 0..4095
  const int t = threadIdx.x;       // 0..63
  const int u  = inter[2 * b];
  const int it = inter[2 * b + 1];
  const float v = (t < EMB_DIM) ? users_emb[u * EMB_DIM + t]
                                : items_emb[it * EMB_DIM + (t - EMB_DIM)];
  embed[b * D_IN + t] = v;
}

// ---------------------------------------------------------------------------
// 2) MLP layer: out = relu(A @ W.T + bias).  A: M x K (row major),
//    W: N x K (row major).  One wave32 per 16x16 output tile.
//    WMMA f32 16x16x4 operand layout (ISA 7.12.2):
//      A lane L: m = L&15, holds K = {2*(L>>4), 2*(L>>4)+1}
//      B lane L: n = L&15, holds K = {2*(L>>4), 2*(L>>4)+1}
//      D reg g, lane L: m = g + 8*(L>>4), n = L&15
// ---------------------------------------------------------------------------
template <int K>
__global__ void mlp_layer_kernel(const float* __restrict__ A,
                                 const float* __restrict__ W,
                                 const float* __restrict__ bias,
                                 float* __restrict__ out,
                                 int N) {
  const int n0   = blockIdx.x * 16;
  const int m0   = blockIdx.y * 16;
  const int lane = threadIdx.x;          // 0..31
  const int half = lane >> 4;
  const int r    = lane & 15;

  v8f acc = {};
#pragma unroll
  for (int k = 0; k < K; k += 4) {
    const int kk = k + 2 * half;
    const v2f a = *(const v2f*)&A[(m0 + r) * K + kk];   // 8B aligned (kk even)
    const v2f b = *(const v2f*)&W[(n0 + r) * K + kk];
    acc = __builtin_amdgcn_wmma_f32_16x16x4_f32(false, a, false, b,
                                                (short)0, acc, false, false);
  }

  const int n = n0 + r;
  const float bb = bias[n];
#pragma unroll
  for (int g = 0; g < 8; ++g) {
    const int m = m0 + g + 8 * half;
    float v = acc[g] + bb;
    out[m * N + n] = v > 0.0f ? v : 0.0f;
  }
}

// ---------------------------------------------------------------------------
// 3) Head: out[b,j] = sigmoid( h2[b,:] . w_head[j,10000:10128]
//                              + w_head[j, items[b]] + b_head[j] )
//    Block = 128 threads (4 waves) -> 64x16 macro tile (M x N).
//    The 16x128 dense w_head tile is staged once into LDS (8 KB) and shared
//    by all 4 waves; K-loop of 32 v_wmma_f32_16x16x4_f32 fully unrolled.
//    Grid: x = m-blocks (fastest) so the 16 gather rows of w_head stay hot
//    in L2 across the whole batch.
// ---------------------------------------------------------------------------
__global__ void head_kernel(const float* __restrict__ H,       // 4096 x 128
                            const float* __restrict__ Wh,      // 10000 x 10128
                            const float* __restrict__ bh,      // 10000
                            const int* __restrict__ inter,     // 4096 x 2
                            float* __restrict__ out) {         // 4096 x 10000
  __shared__ float ldsB[16 * H2];                              // 8 KB

  const int n0 = blockIdx.y * 16;
  const int t  = threadIdx.x;            // 0..127
  const int wave = t >> 5;               // 0..3
  const int lane = t & 31;
  const int half = lane >> 4;
  const int r    = lane & 15;
  const int m0 = (blockIdx.x * 4 + wave) * 16;

  // Stage dense B tile: ldsB[n][k] = Wh[n0+n, 10000+k]; 512 float4 loads.
  {
    const float4* __restrict__ src = (const float4*)Wh;
#pragma unroll
    for (int i = t; i < 16 * H2 / 4; i += 128) {
      const int n  = i >> 5;             // 32 float4 per row
      const int kq = i & 31;
      ((float4*)ldsB)[i] =
          src[((size_t)(n0 + n) * HEAD_IN + NUM_ITEMS) / 4 + kq];
    }
  }
  __syncthreads();

  v8f acc = {};
#pragma unroll
  for (int k = 0; k < H2; k += 4) {
    const int kk = k + 2 * half;
    const v2f a = *(const v2f*)&H[(m0 + r) * H2 + kk];
    const v2f b = *(const v2f*)&ldsB[r * H2 + kk];
    acc = __builtin_amdgcn_wmma_f32_16x16x4_f32(false, a, false, b,
                                                (short)0, acc, false, false);
  }

  const int n = n0 + r;
  const float bb = bh[n];
#pragma unroll
  for (int g = 0; g < 8; ++g) {
    const int m = m0 + g + 8 * half;
    const int item = inter[2 * m + 1];
    const float cross = Wh[(size_t)n * HEAD_IN + item];  // one-hot ≡ gather
    const float x = acc[g] + cross + bb;
    out[(size_t)m * NUM_ITEMS + n] = 1.0f / (1.0f + __expf(-x));
  }
}

// ---------------------------------------------------------------------------
// Launch
// ---------------------------------------------------------------------------
extern "C" void kernel_launch(void* const* d_in, const int* in_sizes, int n_in,
                              void* d_out, int out_size, void* d_ws, size_t ws_size,
                              hipStream_t stream) {
  const int*   inter = (const int*)d_in[0];     // interactions (4096 x 2)
  const float* ue    = (const float*)d_in[1];   // users_emb_w
  const float* ie    = (const float*)d_in[2];   // items_emb_w
  const float* w1    = (const float*)d_in[3];   // 128 x 64
  const float* b1    = (const float*)d_in[4];
  const float* w2    = (const float*)d_in[5];   // 128 x 128
  const float* b2    = (const float*)d_in[6];
  const float* wh    = (const float*)d_in[7];   // 10000 x 10128
  const float* bh    = (const float*)d_in[8];
  float* out = (float*)d_out;

  float* embed = (float*)d_ws;                  // 4096 x 64
  float* h1    = embed + BATCH * D_IN;          // 4096 x 128
  float* h2    = h1 + BATCH * H1;               // 4096 x 128
  // total workspace: 5.25 MB

  gather_embed_kernel<<<BATCH, 64, 0, stream>>>(inter, ue, ie, embed);

  // h1 = relu(embed @ w1.T + b1):  M=4096, N=128, K=64
  mlp_layer_kernel<D_IN><<<dim3(H1 / 16, BATCH / 16), 32, 0, stream>>>(
      embed, w1, b1, h1, H1);

  // h2 = relu(h1 @ w2.T + b2):  M=4096, N=128, K=128
  mlp_layer_kernel<H1><<<dim3(H2 / 16, BATCH / 16), 32, 0, stream>>>(
      h1, w2, b2, h2, H2);

  // out = sigmoid(gather + h2 @ w_head_dense.T + b_head)
  head_kernel<<<dim3(BATCH / 64, NUM_ITEMS / 16), 128, 0, stream>>>(
      h2, wh, bh, inter, out);
}